// MatchBRNN_11175504904491
// MI455X (gfx1250) — compile-verified
//
#include <hip/hip_runtime.h>
#include <cmath>

// Problem constants (match reference)
#define Bsz  16
#define Ssz  256          // seq len == L (memory length)
#define Dsz  256          // input / output feature dim
#define Hsz  128          // hidden per direction
#define DKsz 64           // attention projection dim
#define INsz 512          // SRU input dim = 2*D
#define NLay 2

typedef float v2f __attribute__((ext_vector_type(2)));
typedef float v8f __attribute__((ext_vector_type(8)));
typedef int   v4i __attribute__((vector_size(4 * sizeof(int))));

// async-copy builtin wants: (AS1 v4i*, AS3 v4i*, imm offset, imm cpol)
#define AS_G1(p) ((__attribute__((address_space(1))) v4i*)(p))
#define AS_L3(p) ((__attribute__((address_space(3))) v4i*)(p))

// ---------------------------------------------------------------------------
// One wave computes a 16x16 f32 tile C = A(16xK) @ B(Kx16) [+ bias per col]
// using V_WMMA_F32_16X16X4_F32, K stepped by 4.
// Lane layout (ISA 7.12.2): half = lane>>4, l16 = lane&15.
//   A vgpr v : A[l16][k + 2*half + v]
//   B vgpr v : B[k + 2*half + v][l16]
//   C vgpr r : C[r + 8*half][l16]
// ---------------------------------------------------------------------------
__device__ __forceinline__ void gemm_tile(const float* __restrict__ A, int lda,
                                          const float* __restrict__ Bm, int ldb,
                                          float* __restrict__ C, int ldc,
                                          int K, const float* __restrict__ bias) {
  const int lane = threadIdx.x & 31;
  const int half = lane >> 4;
  const int l16  = lane & 15;
  v8f acc = {};
  for (int k = 0; k < K; k += 4) {
    const int ka = k + 2 * half;
    if ((k & 31) == 0) {
      __builtin_prefetch(Bm + (size_t)(ka + 32) * ldb + l16, 0, 0);   // global_prefetch_b8
      __builtin_prefetch(A + (size_t)l16 * lda + k + 32, 0, 0);
    }
    v2f a = *(const v2f*)(A + (size_t)l16 * lda + ka);
    v2f b;
    b.x = Bm[(size_t)(ka + 0) * ldb + l16];
    b.y = Bm[(size_t)(ka + 1) * ldb + l16];
    acc = __builtin_amdgcn_wmma_f32_16x16x4_f32(false, a, false, b,
                                                (short)0, acc, false, false);
  }
  const float bb = bias ? bias[l16] : 0.0f;
#pragma unroll
  for (int r = 0; r < 8; ++r)
    C[(size_t)(r + 8 * half) * ldc + l16] = acc[r] + bb;
}

// ---------------------------------------------------------------------------
// out0[(s*B+b)*D + d] = x[(b*S+s)*D + d]   (float4 vectorized)
// ---------------------------------------------------------------------------
__global__ void transpose_in_kernel(const float* __restrict__ x,
                                    float* __restrict__ out0) {
  int idx = blockIdx.x * blockDim.x + threadIdx.x;      // over B*S*(D/4)
  if (idx >= Bsz * Ssz * (Dsz / 4)) return;
  int d4   = idx & (Dsz / 4 - 1);
  int rest = idx >> 6;                                   // b*S + s
  int s = rest & (Ssz - 1);
  int b = rest >> 8;
  ((float4*)out0)[((size_t)s * Bsz + b) * (Dsz / 4) + d4] = ((const float4*)x)[idx];
}

// d_out[(b*S+s)*D + d] = out[(s*B+b)*D + d]
__global__ void transpose_out_kernel(const float* __restrict__ out,
                                     float* __restrict__ y) {
  int idx = blockIdx.x * blockDim.x + threadIdx.x;      // over B*S*(D/4), output order
  if (idx >= Bsz * Ssz * (Dsz / 4)) return;
  int d4   = idx & (Dsz / 4 - 1);
  int rest = idx >> 6;
  int s = rest & (Ssz - 1);
  int b = rest >> 8;
  ((float4*)y)[idx] = ((const float4*)out)[((size_t)s * Bsz + b) * (Dsz / 4) + d4];
}

// inputs[row][0:256] = out[row][:]   (row = s*B+b)
__global__ void copy_in_kernel(const float* __restrict__ out,
                               float* __restrict__ inputs) {
  int idx = blockIdx.x * blockDim.x + threadIdx.x;      // over S*B*(D/4)
  if (idx >= Ssz * Bsz * (Dsz / 4)) return;
  int d4  = idx & 63;
  int row = idx >> 6;
  ((float4*)inputs)[(size_t)row * (INsz / 4) + d4] = ((const float4*)out)[idx];
}

// ---------------------------------------------------------------------------
// xt(B,L,DK) = memory(B,L,D) @ w1[actions[b]](D,DK) + b1[actions[b]]
// grid: (16 Mtiles * 4 Ntiles, B), block: 32
// ---------------------------------------------------------------------------
__global__ __launch_bounds__(32)
void xt_kernel(const float* __restrict__ x, const int* __restrict__ actions,
               const float* __restrict__ w1, const float* __restrict__ b1,
               float* __restrict__ xt) {
  int b  = blockIdx.y;
  int tm = blockIdx.x >> 2, tn = blockIdx.x & 3;
  int a  = actions[b];
  gemm_tile(x + (size_t)b * Ssz * Dsz + (size_t)tm * 16 * Dsz, Dsz,
            w1 + (size_t)a * Dsz * DKsz + tn * 16, DKsz,
            xt + (size_t)b * Ssz * DKsz + (size_t)tm * 16 * DKsz + tn * 16, DKsz,
            Dsz, b1 + a * DKsz + tn * 16);
}

// yt(S,B,DK) = out(S,B,D) @ w2[actions[b]](D,DK) + b2[actions[b]]
__global__ __launch_bounds__(32)
void yt_kernel(const float* __restrict__ out, const int* __restrict__ actions,
               const float* __restrict__ w2, const float* __restrict__ b2,
               float* __restrict__ yt) {
  int b  = blockIdx.y;
  int tm = blockIdx.x >> 2, tn = blockIdx.x & 3;
  int a  = actions[b];
  gemm_tile(out + (size_t)tm * 16 * Bsz * Dsz + (size_t)b * Dsz, Bsz * Dsz,
            w2 + (size_t)a * Dsz * DKsz + tn * 16, DKsz,
            yt + (size_t)tm * 16 * Bsz * DKsz + (size_t)b * DKsz + tn * 16, Bsz * DKsz,
            Dsz, b2 + a * DKsz + tn * 16);
}

// ---------------------------------------------------------------------------
// Fused attention: scores -> softmax -> WMMA pooling, per (b, 16-row s tile).
// Writes pools into inputs[:, 256:512].
// grid: (S/16, B), block: 256 (8 waves)
// ---------------------------------------------------------------------------
__global__ __launch_bounds__(256)
void attn_kernel(const float* __restrict__ xt,     // (B,L,DK)
                 const float* __restrict__ yt,     // (S,B,DK)
                 const float* __restrict__ v,      // (A,DK)
                 const int* __restrict__ actions,
                 const unsigned char* __restrict__ x_mask, // (B,L)
                 const float* __restrict__ memory, // x (B,L,D)
                 float* __restrict__ inputs) {     // (S,B,512)
  __shared__ float att[16 * Ssz];   // 16 s rows x 256 l  (16 KB)
  __shared__ float yts[16 * DKsz];  // 4 KB
  __shared__ float vs[DKsz];
  const int b  = blockIdx.y;
  const int s0 = blockIdx.x * 16;
  const int t  = threadIdx.x;
  const int a  = actions[b];

  if (t < DKsz) vs[t] = v[a * DKsz + t];
  for (int e = t; e < 16 * DKsz; e += 256) {
    int sl = e >> 6, k = e & 63;
    yts[e] = yt[(size_t)(s0 + sl) * Bsz * DKsz + (size_t)b * DKsz + k];
  }
  __syncthreads();

  // scores[sl][l] = sum_k vs[k] * tanh(xt[b,l,k] + yts[sl,k]), masked
  const float* xtb = xt + (size_t)b * Ssz * DKsz;
  for (int e = t; e < 16 * Ssz; e += 256) {
    int sl = e >> 8, l = e & 255;
    const float* xr = xtb + (size_t)l * DKsz;
    const float* yr = yts + sl * DKsz;
    float sum = 0.0f;
#pragma unroll 8
    for (int k = 0; k < DKsz; ++k)
      sum += vs[k] * tanhf(xr[k] + yr[k]);
    if (x_mask[b * Ssz + l]) sum = -INFINITY;
    att[sl * Ssz + l] = sum;
  }
  __syncthreads();

  // per-row softmax (16 rows, one thread each — tiny)
  if (t < 16) {
    float* row = att + t * Ssz;
    float m = -INFINITY;
    for (int l = 0; l < Ssz; ++l) m = fmaxf(m, row[l]);
    float ssum = 0.0f;
    for (int l = 0; l < Ssz; ++l) { float e = __expf(row[l] - m); row[l] = e; ssum += e; }
    float inv = 1.0f / ssum;
    for (int l = 0; l < Ssz; ++l) row[l] *= inv;
  }
  __syncthreads();

  // pools(16 x 256) = att(16x256, LDS) @ memory[b](256x256) via WMMA
  const int wave = t >> 5, lane = t & 31;
  const int half = lane >> 4, l16 = lane & 15;
  const float* Bm = memory + (size_t)b * Ssz * Dsz;
  for (int nt = wave; nt < Dsz / 16; nt += 8) {
    v8f acc = {};
    for (int k = 0; k < Ssz; k += 4) {
      const int ka = k + 2 * half;
      v2f av = *(const v2f*)(att + l16 * Ssz + ka);       // ds_load_b64
      v2f bv;
      bv.x = Bm[(size_t)(ka + 0) * Dsz + nt * 16 + l16];
      bv.y = Bm[(size_t)(ka + 1) * Dsz + nt * 16 + l16];
      acc = __builtin_amdgcn_wmma_f32_16x16x4_f32(false, av, false, bv,
                                                  (short)0, acc, false, false);
    }
#pragma unroll
    for (int r = 0; r < 8; ++r) {
      int srow = r + 8 * half;
      int col  = nt * 16 + l16;
      inputs[((size_t)(s0 + srow) * Bsz + b) * INsz + Dsz + col] = acc[r];
    }
  }
}

// ---------------------------------------------------------------------------
// SRU projections: U = inputs(4096x512) @ W(512x512), fwd & bwd via blockIdx.y.
// Block = 256 threads (8 waves) computes a 64x32 C tile; A(64x32) and B(32x32)
// K-chunks are staged into LDS with GLOBAL_LOAD_ASYNC_TO_LDS_B128 (ASYNCcnt),
// then each wave runs a 16x16 WMMA subtile fed from LDS.
// grid: ((4096/64)*(512/32)=1024, 2), block: 256
// ---------------------------------------------------------------------------
#define GBM 64
#define GBN 32
#define GBK 32

__global__ __launch_bounds__(256)
void sru_gemm_kernel(const float* __restrict__ inputs,
                     const float* __restrict__ w_f, const float* __restrict__ w_b,
                     float* __restrict__ U_f, float* __restrict__ U_b) {
  __shared__ float As[GBM * GBK];   // 8 KB row-major 64x32
  __shared__ float Bs[GBK * GBN];   // 4 KB row-major 32x32
  const float* W = blockIdx.y ? w_b : w_f;
  float*       U = blockIdx.y ? U_b : U_f;
  const int m0 = (blockIdx.x >> 4) * GBM;   // 64 M-blocks
  const int n0 = (blockIdx.x & 15) * GBN;   // 16 N-blocks
  const int t    = threadIdx.x;
  const int wave = t >> 5, lane = t & 31;
  const int half = lane >> 4, l16 = lane & 15;
  const int subM = wave >> 1, subN = wave & 1;   // 4 x 2 subtiles of 16x16

  v8f acc = {};
  for (int kk = 0; kk < INsz; kk += GBK) {
    // stage B chunk: 32x32 floats = 256 float4, one per thread
    {
      int row = t >> 3, c4 = (t & 7) << 2;
      __builtin_amdgcn_global_load_async_to_lds_b128(
          AS_G1(W + (size_t)(kk + row) * (4 * Hsz) + n0 + c4),
          AS_L3(Bs + row * GBN + c4), 0, 0);
    }
    // stage A chunk: 64x32 floats = 512 float4, two per thread
#pragma unroll
    for (int i = 0; i < 2; ++i) {
      int fa = t + i * 256;
      int ra = fa >> 3, ca = (fa & 7) << 2;
      __builtin_amdgcn_global_load_async_to_lds_b128(
          AS_G1(inputs + (size_t)(m0 + ra) * INsz + kk + ca),
          AS_L3(As + ra * GBK + ca), 0, 0);
    }
    asm volatile("s_wait_asynccnt 0x0" ::: "memory");
    __syncthreads();
#pragma unroll
    for (int k = 0; k < GBK; k += 4) {
      const int ka = k + 2 * half;
      v2f a = *(const v2f*)(As + (subM * 16 + l16) * GBK + ka);
      v2f b;
      b.x = Bs[(ka + 0) * GBN + subN * 16 + l16];
      b.y = Bs[(ka + 1) * GBN + subN * 16 + l16];
      acc = __builtin_amdgcn_wmma_f32_16x16x4_f32(false, a, false, b,
                                                  (short)0, acc, false, false);
    }
    __syncthreads();
  }
  float* C = U + (size_t)(m0 + subM * 16) * (4 * Hsz) + n0 + subN * 16;
#pragma unroll
  for (int r = 0; r < 8; ++r)
    C[(size_t)(r + 8 * half) * (4 * Hsz) + l16] = acc[r];
}

// ---------------------------------------------------------------------------
// SRU recurrence (both directions are forward scans, as in the reference).
// 4096 threads: dir(2) x B(16) x H(128); serial over S.
// ---------------------------------------------------------------------------
__global__ __launch_bounds__(256)
void sru_scan_kernel(const float* __restrict__ U_f, const float* __restrict__ U_b,
                     const float* __restrict__ bias_f, const float* __restrict__ bias_b,
                     float* __restrict__ out_next) {  // (S,B,D)
  int t = blockIdx.x * blockDim.x + threadIdx.x;
  if (t >= 2 * Bsz * Hsz) return;
  const int dir = t >> 11;
  const int bh  = t & 2047;
  const int b   = bh >> 7;
  const int h   = bh & (Hsz - 1);
  const float* U    = dir ? U_b : U_f;
  const float* bias = dir ? bias_b : bias_f;
  const float bf = bias[h], br = bias[Hsz + h];
  float c = 0.0f;
  for (int s = 0; s < Ssz; ++s) {
    const float* u = U + ((size_t)s * Bsz + b) * (4 * Hsz);
    const float u0 = u[h];
    const float uf = u[Hsz + h];
    const float ur = u[2 * Hsz + h];
    const float hw = u[3 * Hsz + h];
    const float f = 1.0f / (1.0f + __expf(-(uf + bf)));
    const float r = 1.0f / (1.0f + __expf(-(ur + br)));
    c = f * c + (1.0f - f) * u0;
    const float hval = r * tanhf(c) + (1.0f - r) * hw;
    out_next[((size_t)s * Bsz + b) * Dsz + dir * Hsz + h] = hval;
  }
}

// ---------------------------------------------------------------------------
extern "C" void kernel_launch(void* const* d_in, const int* in_sizes, int n_in,
                              void* d_out, int out_size, void* d_ws, size_t ws_size,
                              hipStream_t stream) {
  const float*         x       = (const float*)d_in[0];
  const unsigned char* x_mask  = (const unsigned char*)d_in[1];
  const int*           actions = (const int*)d_in[2];
  const float*         w1      = (const float*)d_in[3];
  const float*         b1      = (const float*)d_in[4];
  const float*         w2      = (const float*)d_in[5];
  const float*         b2      = (const float*)d_in[6];
  const float*         v       = (const float*)d_in[7];
  const float*         sru_w_f = (const float*)d_in[8];
  const float*         sru_b_f = (const float*)d_in[9];
  const float*         sru_w_b = (const float*)d_in[10];
  const float*         sru_b_b = (const float*)d_in[11];
  float*               y       = (float*)d_out;

  // workspace layout (floats)
  float* ws     = (float*)d_ws;
  float* out0   = ws;                         // S*B*D   = 1,048,576
  float* out1   = out0 + (size_t)Ssz * Bsz * Dsz;
  float* xt     = out1 + (size_t)Ssz * Bsz * Dsz;     // B*S*DK = 262,144
  float* yt     = xt + (size_t)Bsz * Ssz * DKsz;      // S*B*DK
  float* inputs = yt + (size_t)Ssz * Bsz * DKsz;      // S*B*512 = 2,097,152
  float* U_f    = inputs + (size_t)Ssz * Bsz * INsz;  // S*B*512
  float* U_b    = U_f + (size_t)Ssz * Bsz * INsz;     // S*B*512

  const int nElem4 = Bsz * Ssz * Dsz / 4;

  // out0 = transpose(x) -> (S,B,D)
  transpose_in_kernel<<<(nElem4 + 255) / 256, 256, 0, stream>>>(x, out0);

  // xt depends only on x and w1[action]; identical across layers -> compute once
  xt_kernel<<<dim3(64, Bsz), 32, 0, stream>>>(x, actions, w1, b1, xt);

  float* cur = out0;
  float* nxt = out1;
  for (int layer = 0; layer < NLay; ++layer) {
    yt_kernel<<<dim3(64, Bsz), 32, 0, stream>>>(cur, actions, w2, b2, yt);
    copy_in_kernel<<<(nElem4 + 255) / 256, 256, 0, stream>>>(cur, inputs);
    attn_kernel<<<dim3(Ssz / 16, Bsz), 256, 0, stream>>>(
        xt, yt, v, actions, x_mask, x, inputs);
    sru_gemm_kernel<<<dim3((Ssz * Bsz / GBM) * (INsz / GBN), 2), 256, 0, stream>>>(
        inputs,
        sru_w_f + (size_t)layer * INsz * 4 * Hsz,
        sru_w_b + (size_t)layer * INsz * 4 * Hsz,
        U_f, U_b);
    sru_scan_kernel<<<(2 * Bsz * Hsz + 255) / 256, 256, 0, stream>>>(
        U_f, U_b,
        sru_b_f + (size_t)layer * 2 * Hsz,
        sru_b_b + (size_t)layer * 2 * Hsz,
        nxt);
    float* tmp = cur; cur = nxt; nxt = tmp;
  }

  // cur now holds final (S,B,D); transpose to (B,S,D)
  transpose_out_kernel<<<(nElem4 + 255) / 256, 256, 0, stream>>>(cur, y);
}